// GNNModel_49752901157156
// MI455X (gfx1250) — compile-verified
//
#include <hip/hip_runtime.h>
#include <hip/hip_bf16.h>

typedef __attribute__((ext_vector_type(2))) float v2f;
typedef __attribute__((ext_vector_type(8))) float v8f;
typedef __attribute__((ext_vector_type(4))) unsigned int v4u;
typedef __attribute__((ext_vector_type(8))) int v8i;
typedef __attribute__((ext_vector_type(4))) int v4i;

#define HDIM 128

// ---------------------------------------------------------------------------
// TDM helper: 1-D contiguous DMA of `nelem` dwords global -> LDS, with
// optional LDS padding (pad_amount dwords inserted every pad_interval dwords)
// for bank-conflict-free fragment reads.  D# layout per CDNA5 ISA ch.8.
// This toolchain exposes the 6-arg builtin form:
//   (v4u group0, v8i group1, v4i group2, v4i group3, v8i extra, i32 cpol)
// ---------------------------------------------------------------------------
__device__ __forceinline__ unsigned lds_off(const void* p) {
    // generic pointer to LDS: low 32 bits hold the LDS byte offset
    return (unsigned)(uintptr_t)p;
}

__device__ __forceinline__ void tdm_load_1d(unsigned lds_byte,
                                            const void* gptr,
                                            unsigned nelem,      // dwords
                                            unsigned icode,      // pad interval code
                                            unsigned acode,      // pad amount code
                                            bool pad_en) {
    unsigned long long ga = (unsigned long long)(uintptr_t)gptr;
    v4u g0;
    g0[0] = 1u;                                            // count=1, user mode
    g0[1] = lds_byte;                                      // lds_addr
    g0[2] = (unsigned)(ga & 0xffffffffu);                  // global_addr[31:0]
    g0[3] = (unsigned)((ga >> 32) & 0x01ffffffu) | (2u << 30);  // addr[56:32]|type=2

    unsigned d0 = (2u << 16);                              // data_size = 4B
    if (pad_en) d0 |= (1u << 20) | (icode << 22) | (acode << 25);
    v8i g1;
    g1[0] = (int)d0;                                       // mask=0, flags
    g1[1] = (int)((nelem & 0xffffu) << 16);                // tensor_dim0 lo
    g1[2] = (int)(((nelem >> 16) & 0xffffu) | (1u << 16)); // tensor_dim0 hi | tensor_dim1=1
    g1[3] = (int)((nelem & 0xffffu) << 16);                // tile_dim0 = nelem (<=65535)
    g1[4] = 0;                                             // tile_dim1=0 (1-D), tile_dim2=0
    g1[5] = (int)nelem;                                    // dim0 stride (unused for 1 row)
    g1[6] = 0;
    g1[7] = 0;
    v4i g2 = {0, 0, 0, 0};
    v4i g3 = {0, 0, 0, 0};
    v8i g4 = {0, 0, 0, 0, 0, 0, 0, 0};
    __builtin_amdgcn_tensor_load_to_lds(g0, g1, g2, g3, g4, 0);
}

// ---------------------------------------------------------------------------
// Init: agg = feat (self-loop contribution), deg = 1 (self loop)
// ---------------------------------------------------------------------------
__global__ void copy_init_kernel(const float* __restrict__ feat,
                                 float* __restrict__ agg,
                                 float* __restrict__ deg, int N) {
    int i = blockIdx.x * blockDim.x + threadIdx.x;
    int n4 = N * (HDIM / 4);
    if (i < n4) ((float4*)agg)[i] = ((const float4*)feat)[i];
    if (deg != nullptr && i < N) deg[i] = 1.0f;
}

__global__ void edge_deg_kernel(const long long* __restrict__ dst,
                                float* __restrict__ deg, int E) {
    int e = blockIdx.x * blockDim.x + threadIdx.x;
    if (e < E) atomicAdd(&deg[(int)dst[e]], 1.0f);
}

__global__ void recip_kernel(const float* __restrict__ deg,
                             float* __restrict__ deginv, int N) {
    int i = blockIdx.x * blockDim.x + threadIdx.x;
    if (i < N) deginv[i] = 1.0f / deg[i];   // deg >= 1 due to self loop
}

// ---------------------------------------------------------------------------
// Scatter-add: one wave32 per edge; each lane moves one float4 (32*16B = 512B)
// ---------------------------------------------------------------------------
__global__ void edge_scatter_kernel(const float* __restrict__ feat,
                                    const long long* __restrict__ src,
                                    const long long* __restrict__ dst,
                                    float* __restrict__ agg, int E) {
    int gid = blockIdx.x * blockDim.x + threadIdx.x;
    int e = gid >> 5;
    int lane = gid & 31;
    if (e >= E) return;
    int s = (int)src[e];
    int d = (int)dst[e];
    float4 v = ((const float4*)(feat + (size_t)s * HDIM))[lane];
    float* ap = agg + (size_t)d * HDIM + lane * 4;
    atomicAdd(ap + 0, v.x);
    atomicAdd(ap + 1, v.y);
    atomicAdd(ap + 2, v.z);
    atomicAdd(ap + 3, v.w);
}

// ---------------------------------------------------------------------------
// Fused SAGE GEMM, TDM-staged operands, WMMA f32 16x16x4:
//   out[m,n] = act( sum_k A2[m,k]*B2[k,n]
//                 + (HAS_A1 ? deginv[m]*sum_k A1[m,k]*B1[k,n] : 0) + bias[n] )
// Block = (32,8) = 8 waves covering 64 rows x NC cols.
//   wave wy: row tile ry = (wy&3)*16, col group cg = wy>>2 (TC=NC/32 tiles).
// LDS staging via 4 TDM 1-D descriptors (wave 0), padded for conflict-free
// ds reads:  A stride 132 dw (pad 4/128), B stride 136 dw (pad 8/128) or
// 72 dw (pad 8/64) for NC=64.
// Fragment layouts per CDNA5 ISA 7.12.2 (wave32):
//   A 16x4 : a[v] = A[M=l16][K = v + 2*h]     -> one ds_load_b64
//   B 4x16 : b[v] = B[K = v + 2*h][N = l16]
//   C 16x16: d[v] = D[M = v + 8*h][N = l16]
// ---------------------------------------------------------------------------
template <int NC, bool HAS_A1, bool RELU>
__global__ __launch_bounds__(256) void sage_gemm_tdm(
    const float* __restrict__ A2, const float* __restrict__ B2,
    const float* __restrict__ A1, const float* __restrict__ B1,
    const float* __restrict__ deginv, const float* __restrict__ bias,
    float* __restrict__ out, int M) {
    constexpr int TC = NC / 32;                  // col tiles per wave
    constexpr int AS = 132;                      // padded A row stride (dwords)
    constexpr int BS = (NC == 128) ? 136 : 72;   // padded B row stride (dwords)
    constexpr unsigned BPI = (NC == 128) ? 6u : 5u;  // pad interval code 128/64 dw

    __shared__ float sB2[128 * BS];
    __shared__ float sB1[HAS_A1 ? 128 * BS : 1];
    __shared__ float sA2[64 * AS];
    __shared__ float sA1[HAS_A1 ? 64 * AS : 1];

    const int lane = threadIdx.x;            // 0..31
    const int wy   = threadIdx.y;            // 0..7
    const int l16  = lane & 15;
    const int hlf  = lane >> 4;              // 0 or 1
    const int row0 = blockIdx.x * 64;
    const int ry   = (wy & 3) * 16;
    const int colw = (wy >> 2) * (NC / 2);

    if (wy == 0) {
        int rem = M - row0;
        unsigned rows = (unsigned)(rem < 64 ? rem : 64);
        unsigned nA = rows * HDIM;
        tdm_load_1d(lds_off(sB2), B2, 128 * NC, BPI, 7u, true);
        if (HAS_A1) tdm_load_1d(lds_off(sB1), B1, 128 * NC, BPI, 7u, true);
        tdm_load_1d(lds_off(sA2), A2 + (size_t)row0 * HDIM, nA, 6u, 3u, true);
        if (HAS_A1)
            tdm_load_1d(lds_off(sA1), A1 + (size_t)row0 * HDIM, nA, 6u, 3u, true);
        __builtin_amdgcn_s_wait_tensorcnt(0);
    }
    __syncthreads();

    int arow = row0 + ry + l16;
    if (arow >= M) arow = M - 1;             // clamp; EXEC stays all-ones for WMMA
    const float s1 = HAS_A1 ? deginv[arow] : 0.0f;

    const float* aP2 = sA2 + (ry + l16) * AS;
    const float* aP1 = HAS_A1 ? (sA1 + (ry + l16) * AS) : nullptr;

    v8f acc[TC];
#pragma unroll
    for (int t = 0; t < TC; ++t) acc[t] = (v8f){0, 0, 0, 0, 0, 0, 0, 0};

#pragma unroll 2
    for (int k0 = 0; k0 < HDIM; k0 += 4) {
        const int ka = k0 + 2 * hlf;
        v2f a2 = *(const v2f*)(aP2 + ka);            // ds_load_b64
#pragma unroll
        for (int t = 0; t < TC; ++t) {
            const int col = colw + t * 16 + l16;
            v2f b2;
            b2.x = sB2[ka * BS + col];
            b2.y = sB2[(ka + 1) * BS + col];
            acc[t] = __builtin_amdgcn_wmma_f32_16x16x4_f32(
                false, a2, false, b2, (short)0, acc[t], false, false);
        }
        if (HAS_A1) {
            v2f a1 = *(const v2f*)(aP1 + ka);
            a1 = a1 * s1;
#pragma unroll
            for (int t = 0; t < TC; ++t) {
                const int col = colw + t * 16 + l16;
                v2f b1;
                b1.x = sB1[ka * BS + col];
                b1.y = sB1[(ka + 1) * BS + col];
                acc[t] = __builtin_amdgcn_wmma_f32_16x16x4_f32(
                    false, a1, false, b1, (short)0, acc[t], false, false);
            }
        }
    }

#pragma unroll
    for (int t = 0; t < TC; ++t) {
        const int col = colw + t * 16 + l16;
        const float bv = bias[col];
#pragma unroll
        for (int v = 0; v < 8; ++v) {
            const int m = row0 + ry + v + 8 * hlf;
            if (m < M) {
                float r = acc[t][v] + bv;
                if (RELU) r = fmaxf(r, 0.0f);
                out[(size_t)m * NC + col] = r;
            }
        }
    }
}

// ---------------------------------------------------------------------------
// GraphNorm: two-pass column reduction (128 cols), then apply + relu
// ---------------------------------------------------------------------------
__global__ void zero_kernel(float* __restrict__ p, int n) {
    int i = blockIdx.x * blockDim.x + threadIdx.x;
    if (i < n) p[i] = 0.0f;
}

__global__ void col_reduce_kernel(const float* __restrict__ h,
                                  const float* __restrict__ sums,
                                  const float* __restrict__ ms,
                                  float* __restrict__ out128, int M,
                                  float invM) {
    __shared__ float sm[256];
    const int tid = threadIdx.x;
    const int c = tid & (HDIM - 1);
    const int rstart = blockIdx.x * 2 + (tid >> 7);
    float center = 0.0f;
    if (sums != nullptr) center = sums[c] * invM * ms[c];
    float acc = 0.0f;
    for (int r = rstart; r < M; r += gridDim.x * 2) {
        float v = h[(size_t)r * HDIM + c];
        if (sums != nullptr) {
            v -= center;
            v = v * v;
        }
        acc += v;
    }
    sm[tid] = acc;
    __syncthreads();
    if (tid < HDIM) atomicAdd(&out128[c], sm[tid] + sm[tid + HDIM]);
}

__global__ void gn_apply_kernel(float* __restrict__ h,
                                const float* __restrict__ sums,
                                const float* __restrict__ sumsq,
                                const float* __restrict__ w,
                                const float* __restrict__ b,
                                const float* __restrict__ ms, int M,
                                float invM) {
    int i = blockIdx.x * blockDim.x + threadIdx.x;
    if (i >= M * HDIM) return;
    const int c = i & (HDIM - 1);
    const float mean = sums[c] * invM;
    const float var = sumsq[c] * invM;
    const float v = h[i] - mean * ms[c];
    const float r = w[c] * v * rsqrtf(var + 1e-5f) + b[c];
    h[i] = fmaxf(r, 0.0f);
}

// ---------------------------------------------------------------------------
extern "C" void kernel_launch(void* const* d_in, const int* in_sizes, int n_in,
                              void* d_out, int out_size, void* d_ws,
                              size_t ws_size, hipStream_t stream) {
    const float* x       = (const float*)d_in[0];
    const long long* ei  = (const long long*)d_in[1];
    const float* w1_l    = (const float*)d_in[2];
    const float* b1_l    = (const float*)d_in[3];
    const float* w1_r    = (const float*)d_in[4];
    const float* w2_l    = (const float*)d_in[5];
    const float* b2_l    = (const float*)d_in[6];
    const float* w2_r    = (const float*)d_in[7];
    const float* gn1_w   = (const float*)d_in[8];
    const float* gn1_b   = (const float*)d_in[9];
    const float* gn1_ms  = (const float*)d_in[10];
    const float* gn2_w   = (const float*)d_in[11];
    const float* gn2_b   = (const float*)d_in[12];
    const float* gn2_ms  = (const float*)d_in[13];
    const float* lin_w   = (const float*)d_in[14];
    const float* lin_b   = (const float*)d_in[15];
    float* out = (float*)d_out;

    const int N = in_sizes[0] / HDIM;
    const int E = in_sizes[1] / 2;
    const long long* src = ei;
    const long long* dst = ei + E;
    const float invM = 1.0f / (float)N;

    // workspace carve-up (~77.5 MB)
    float* ws     = (float*)d_ws;
    float* deg    = ws;                          // N
    float* deginv = deg + N;                     // N
    float* sums   = deginv + N;                  // 128
    float* sumsq  = sums + 128;                  // 128
    float* agg    = sumsq + 128;                 // N*128
    float* h1     = agg + (size_t)N * HDIM;      // N*128
    float* h2     = h1 + (size_t)N * HDIM;       // N*128

    const int TB = 256;
    dim3 gblk(32, 8);
    int rowBlocks = (N + 63) / 64;

    // --- degree + deg_inv --------------------------------------------------
    copy_init_kernel<<<(N * 32 + TB - 1) / TB, TB, 0, stream>>>(x, agg, deg, N);
    edge_deg_kernel<<<(E + TB - 1) / TB, TB, 0, stream>>>(dst, deg, E);
    recip_kernel<<<(N + TB - 1) / TB, TB, 0, stream>>>(deg, deginv, N);

    // --- layer 1: scatter + fused dual GEMM + bias + relu ------------------
    edge_scatter_kernel<<<(E * 32 + TB - 1) / TB, TB, 0, stream>>>(x, src, dst, agg, E);
    sage_gemm_tdm<128, true, true><<<rowBlocks, gblk, 0, stream>>>(
        x, w1_r, agg, w1_l, deginv, b1_l, h1, N);

    // --- GraphNorm 1 + relu (in place on h1) -------------------------------
    zero_kernel<<<1, 256, 0, stream>>>(sums, 256);
    col_reduce_kernel<<<512, 256, 0, stream>>>(h1, nullptr, nullptr, sums, N, invM);
    col_reduce_kernel<<<512, 256, 0, stream>>>(h1, sums, gn1_ms, sumsq, N, invM);
    gn_apply_kernel<<<(N * HDIM + TB - 1) / TB, TB, 0, stream>>>(
        h1, sums, sumsq, gn1_w, gn1_b, gn1_ms, N, invM);

    // --- layer 2 -----------------------------------------------------------
    copy_init_kernel<<<(N * 32 + TB - 1) / TB, TB, 0, stream>>>(h1, agg, nullptr, N);
    edge_scatter_kernel<<<(E * 32 + TB - 1) / TB, TB, 0, stream>>>(h1, src, dst, agg, E);
    sage_gemm_tdm<128, true, true><<<rowBlocks, gblk, 0, stream>>>(
        h1, w2_r, agg, w2_l, deginv, b2_l, h2, N);

    // --- GraphNorm 2 + relu (in place on h2) -------------------------------
    zero_kernel<<<1, 256, 0, stream>>>(sums, 256);
    col_reduce_kernel<<<512, 256, 0, stream>>>(h2, nullptr, nullptr, sums, N, invM);
    col_reduce_kernel<<<512, 256, 0, stream>>>(h2, sums, gn2_ms, sumsq, N, invM);
    gn_apply_kernel<<<(N * HDIM + TB - 1) / TB, TB, 0, stream>>>(
        h2, sums, sumsq, gn2_w, gn2_b, gn2_ms, N, invM);

    // --- output head: [N,128] @ [128,64] + bias (no relu) ------------------
    sage_gemm_tdm<64, false, false><<<rowBlocks, gblk, 0, stream>>>(
        h2, lin_w, nullptr, nullptr, nullptr, lin_b, out, N);
}